// EdgeConv_48086453846655
// MI455X (gfx1250) — compile-verified
//
#include <hip/hip_runtime.h>
#include <hip/hip_bf16.h>
#include <math.h>

typedef __attribute__((ext_vector_type(2))) float v2f;
typedef __attribute__((ext_vector_type(8))) float v8f;
typedef __attribute__((ext_vector_type(4))) int   v4i;

#define B_ 2
#define C_ 64
#define N_ 8192
#define O_ 64
#define K_ 20
#define F_ 128
#define EPS_ 1e-5f
#define SLOPE_ 0.2f

// ---- async-to-LDS availability (guarded; sync fallback otherwise) ----
#if defined(__has_builtin)
# if __has_builtin(__builtin_amdgcn_global_load_async_to_lds_b128) && __has_builtin(__builtin_amdgcn_s_wait_asynccnt)
#  define HAVE_ASYNC 1
# endif
#endif
#ifndef HAVE_ASYNC
# define HAVE_ASYNC 0
#endif
#define AS1 __attribute__((address_space(1)))
#define AS3 __attribute__((address_space(3)))

// ---- workspace layout (bytes) ----
#define WS_IDX_OFF  0ull                                   // int[B*N*K]      = 1.31 MB
#define WS_XT_OFF   (WS_IDX_OFF + (size_t)B_*N_*K_*4)      // float[B*N*C]    = 4 MB
#define WS_SQ_OFF   (WS_XT_OFF + (size_t)B_*N_*C_*4)       // float[B*N]      = 64 KB
#define WS_STAT_OFF (WS_SQ_OFF + (size_t)B_*N_*4)          // float[128] mean,invstd
#define WS_PART_OFF (WS_STAT_OFF + 512ull)                 // float[NBLK*128] block partials

#define NODES 4
#define NBLK  (B_*(N_/NODES))                              // 4096 blocks for k_edge

// ------------------------------------------------------------------
// Kernel 0: transpose x (B,C,N) -> xt (B,N,C), plus squared norms
// ------------------------------------------------------------------
__global__ void k_transpose(const float* __restrict__ x,
                            float* __restrict__ xt,
                            float* __restrict__ sq) {
  int gid = blockIdx.x * blockDim.x + threadIdx.x;   // over B*N
  if (gid >= B_ * N_) return;
  int b = gid / N_, n = gid % N_;
  const float* xp = x + (size_t)b * C_ * N_ + n;
  float* xr = xt + (size_t)gid * C_;
  float acc = 0.f;
  #pragma unroll
  for (int c = 0; c < C_; ++c) {
    float v = xp[(size_t)c * N_];
    acc += v * v;
    xr[c] = v;
  }
  sq[gid] = acc;
}

// ------------------------------------------------------------------
// Kernel 1: KNN via f32 WMMA gram tiles + streaming register top-K
//   block = 256 threads (8 waves), owns 128 rows; cols in chunks of 32,
//   B-chunk double-buffered via async global->LDS DMA.
//   wave w -> col-tile ct=w&1 (B regs reused), row-tiles rt0 and rt0+4.
// ------------------------------------------------------------------
#define ROWS 128
#define CCH  32
#define AST  68   // LDS row stride for feature tiles (bank-conflict-free)
#define DST  34   // LDS row stride for the distance tile

union KnnShared {
  struct {
    float Ar[ROWS * AST];
    float Br[2][CCH * AST];
    float Ds[ROWS * DST];
    float sA[ROWS];
    float sB[CCH];
  } c;
  struct {           // post-loop overlay: per-thread top-K lists for merge
    float d[256 * K_];
    int   i[256 * K_];
  } m;
};

__global__ void __launch_bounds__(256)
k_knn(const float* __restrict__ xt, const float* __restrict__ sqg,
      int* __restrict__ idxb) {
  __shared__ KnnShared sh;

  const int tid   = threadIdx.x;
  const int wave  = tid >> 5;          // wave32
  const int laneM = tid & 15;
  const int hf    = (tid >> 4) & 1;
  const int b     = blockIdx.y;
  const int row0  = blockIdx.x * ROWS;
  const size_t bn = (size_t)b * N_;

  // stage this block's 128 rows (coalesced: contiguous region of xt)
  for (int i = tid; i < ROWS * C_; i += 256) {
    int r = i >> 6, c = i & 63;
    sh.c.Ar[r * AST + c] = xt[(bn + row0) * C_ + i];
  }
  if (tid < ROWS) sh.c.sA[tid] = sqg[bn + row0 + tid];

  float best[K_];
  int   bidx[K_];
  #pragma unroll
  for (int i = 0; i < K_; ++i) { best[i] = -3.0e38f; bidx[i] = 0; }

  // stage one 32x64 column chunk into Br[sel]; 2 x b128 per thread
#if HAVE_ASYNC
  #define STAGE_BR(sel, c0)                                              \
    do {                                                                 \
      const float* gsrc_ = &xt[(bn + (c0)) * C_];                        \
      _Pragma("unroll")                                                  \
      for (int i_ = tid; i_ < CCH * 16; i_ += 256) {                     \
        int r_ = i_ >> 4, q_ = (i_ & 15) << 2;                           \
        __builtin_amdgcn_global_load_async_to_lds_b128(                  \
            (AS1 v4i*)(gsrc_ + r_ * C_ + q_),                           \
            (AS3 v4i*)&sh.c.Br[sel][r_ * AST + q_], 0, 0);               \
      }                                                                  \
    } while (0)
#else
  #define STAGE_BR(sel, c0)                                              \
    do {                                                                 \
      const float* gsrc_ = &xt[(bn + (c0)) * C_];                        \
      _Pragma("unroll")                                                  \
      for (int i_ = tid; i_ < CCH * C_; i_ += 256) {                     \
        int r_ = i_ >> 6, c_ = i_ & 63;                                  \
        sh.c.Br[sel][r_ * AST + c_] = gsrc_[i_];                         \
      }                                                                  \
    } while (0)
#endif

  STAGE_BR(0, 0);   // prime the pipeline

  for (int col0 = 0, it = 0; col0 < N_; col0 += CCH, ++it) {
    const int buf = it & 1, nxt = buf ^ 1;
    const bool more = (col0 + CCH) < N_;
    if (more) STAGE_BR(nxt, col0 + CCH);           // overlap next chunk DMA
    if (tid < CCH) sh.c.sB[tid] = sqg[bn + col0 + tid];
#if HAVE_ASYNC
    if (more) __builtin_amdgcn_s_wait_asynccnt(2); // current chunk landed
    else      __builtin_amdgcn_s_wait_asynccnt(0);
#else
    if (more) __builtin_prefetch(&xt[(bn + col0 + CCH) * C_ + tid * 8], 0, 0);
#endif
    __syncthreads();                               // (A) chunk visible to all

    // this wave's column tile: preload B operands once, reuse for 2 row tiles
    {
      const int ct  = wave & 1;
      const int rt0 = wave >> 1;
      v2f breg[16];
      {
        const float* bp = &sh.c.Br[buf][(ct * 16 + laneM) * AST + 2 * hf];
        #pragma unroll
        for (int s = 0; s < 16; ++s) { breg[s][0] = bp[4 * s]; breg[s][1] = bp[4 * s + 1]; }
      }
      #pragma unroll
      for (int half = 0; half < 2; ++half) {
        const int rt = rt0 + 4 * half;
        v2f areg[16];
        const float* ap = &sh.c.Ar[(rt * 16 + laneM) * AST + 2 * hf];
        #pragma unroll
        for (int s = 0; s < 16; ++s) { areg[s][0] = ap[4 * s]; areg[s][1] = ap[4 * s + 1]; }
        v8f acc = {0.f, 0.f, 0.f, 0.f, 0.f, 0.f, 0.f, 0.f};
        #pragma unroll
        for (int s = 0; s < 16; ++s)   // back-to-back WMMA, operands in regs
          acc = __builtin_amdgcn_wmma_f32_16x16x4_f32(
              false, areg[s], false, breg[s], (short)0, acc, false, false);
        #pragma unroll
        for (int r = 0; r < 8; ++r) {
          int row = rt * 16 + r + 8 * hf;          // D: M = r + 8*(lane>=16)
          int col = ct * 16 + laneM;
          sh.c.Ds[row * DST + col] = 2.f * acc[r] - sh.c.sA[row] - sh.c.sB[col];
        }
      }
    }
    __syncthreads();                               // (B) Ds ready

    // two threads per row: thread scans its 16-column half of the chunk
    {
      const int row = tid & 127, ch = tid >> 7;
      const float* dp = &sh.c.Ds[row * DST + ch * 16];
      const int gbase = col0 + ch * 16;
      for (int j = 0; j < 16; ++j) {
        float d = dp[j];
        if (d > best[K_ - 1]) {
          best[K_ - 1] = d; bidx[K_ - 1] = gbase + j;
          #pragma unroll
          for (int i = K_ - 1; i > 0; --i) {
            if (best[i] > best[i - 1]) {
              float tf = best[i]; best[i] = best[i - 1]; best[i - 1] = tf;
              int   ti = bidx[i]; bidx[i] = bidx[i - 1]; bidx[i - 1] = ti;
            }
          }
        }
      }
    }
    __syncthreads();                               // (C) scan done; bufs free
  }

  // merge the two half-row top-K lists (overlay scratch over tile buffers)
  #pragma unroll
  for (int k = 0; k < K_; ++k) {
    sh.m.d[tid * K_ + k] = best[k];
    sh.m.i[tid * K_ + k] = bidx[k];
  }
  __syncthreads();
  if (tid < ROWS) {
    const float* dA = &sh.m.d[tid * K_];
    const int*   iA = &sh.m.i[tid * K_];
    const float* dB = &sh.m.d[(tid + 128) * K_];
    const int*   iB = &sh.m.i[(tid + 128) * K_];
    int* op = idxb + (bn + row0 + tid) * K_;
    int i = 0, j = 0;
    for (int k = 0; k < K_; ++k) {
      float da = dA[i], db = dB[j];
      int   ia = iA[i], ib = iB[j];
      bool takeA = (da > db) || (da == db && ia < ib);
      op[k] = takeA ? ia : ib;
      if (takeA) ++i; else ++j;
    }
  }
  #undef STAGE_BR
}

// ------------------------------------------------------------------
// Kernels 2/4: edge GEMM via f32 WMMA, 4 independent accumulators.
//   STATS=true : deterministic per-block sum/sumsq partials
//   STATS=false: BN + LeakyReLU + max over K -> output
// block = 64 threads (2 waves), 4 nodes -> 80 rows -> 5 row-tiles of 16
// ------------------------------------------------------------------
#define WST 132
#define EST 132

template <bool STATS>
__global__ void __launch_bounds__(64)
k_edge(const float* __restrict__ xt, const int* __restrict__ idxb,
       const float* __restrict__ W,
       const float* __restrict__ gamma, const float* __restrict__ beta,
       float* __restrict__ bpart, const float* __restrict__ stats,
       float* __restrict__ out) {
  __shared__ float Wl[O_ * WST];          // W staged as [o][f], padded
  __shared__ float Ed[2][16 * EST];       // per-wave edge-row tile (16 x 128)
  __shared__ float partS[2][2][O_];       // deterministic partial sums
  __shared__ float partQ[2][2][O_];
  __shared__ unsigned int mx[NODES * O_]; // encoded max over K
  __shared__ float meanL[O_], invL[O_], gamL[O_], betL[O_];

  const int tid   = threadIdx.x;
  const int wave  = tid >> 5;
  const int laneM = tid & 15;
  const int hf    = (tid >> 4) & 1;
  const int nb    = N_ / NODES;
  const int b     = blockIdx.x / nb;
  const int n0    = (blockIdx.x % nb) * NODES;
  const size_t bn = (size_t)b * N_;

  for (int i = tid; i < O_ * F_; i += 64) {
    int o = i >> 7, f = i & 127;
    Wl[o * WST + f] = W[i];
  }
  if (!STATS) {
    if (tid < O_) {
      meanL[tid] = stats[tid];
      invL[tid]  = stats[64 + tid];
      gamL[tid]  = gamma[tid];
      betL[tid]  = beta[tid];
    }
    for (int i = tid; i < NODES * O_; i += 64) mx[i] = 0u;
  }
  __syncthreads();

  float s1a[4] = {0.f, 0.f, 0.f, 0.f};
  float s2a[4] = {0.f, 0.f, 0.f, 0.f};

  for (int it = 0; it < 3; ++it) {          // 5 row-tiles over 2 waves
    int rt = it * 2 + wave;
    bool active = rt < 5;                   // wave-uniform predicate
    if (active) {
      // stage one edge row-half per lane: edge = [nbr-center | center]
      int lr = rt * 16 + laneM;
      int nl = lr / K_, kk = lr % K_;
      int n = n0 + nl;
      int nbr = idxb[(bn + n) * K_ + kk];
      const float4* cen4 = (const float4*)&xt[(bn + n) * C_];
      const float4* nb4  = (const float4*)&xt[(bn + nbr) * C_];
      float4* er4 = (float4*)&Ed[wave][laneM * EST + hf * C_];
      if (hf == 0) {
        #pragma unroll
        for (int f = 0; f < 16; ++f) {
          float4 a = nb4[f], c = cen4[f];
          er4[f] = make_float4(a.x - c.x, a.y - c.y, a.z - c.z, a.w - c.w);
        }
      } else {
        #pragma unroll
        for (int f = 0; f < 16; ++f) er4[f] = cen4[f];
      }
    }
    __syncthreads();
    if (active) {
      v8f acc[4];
      #pragma unroll
      for (int ct = 0; ct < 4; ++ct) {
        v8f z = {0.f, 0.f, 0.f, 0.f, 0.f, 0.f, 0.f, 0.f};
        acc[ct] = z;
      }
      const float* ap = &Ed[wave][laneM * EST + 2 * hf];
      const float* bp = &Wl[laneM * WST + 2 * hf];
      #pragma unroll
      for (int s = 0; s < 32; ++s) {        // F=128 in steps of 4
        v2f a;
        a[0] = ap[4 * s]; a[1] = ap[4 * s + 1];
        #pragma unroll
        for (int ct = 0; ct < 4; ++ct) {    // 4 independent WMMA per k-step
          v2f bb;
          bb[0] = bp[ct * 16 * WST + 4 * s];
          bb[1] = bp[ct * 16 * WST + 4 * s + 1];
          acc[ct] = __builtin_amdgcn_wmma_f32_16x16x4_f32(
              false, a, false, bb, (short)0, acc[ct], false, false);
        }
      }
      if (STATS) {
        #pragma unroll
        for (int ct = 0; ct < 4; ++ct) {
          float s1 = 0.f, s2 = 0.f;
          #pragma unroll
          for (int r = 0; r < 8; ++r) { float h = acc[ct][r]; s1 += h; s2 += h * h; }
          s1a[ct] += s1;
          s2a[ct] += s2;
        }
      } else {
        #pragma unroll
        for (int ct = 0; ct < 4; ++ct) {
          int o = ct * 16 + laneM;
          float mean = meanL[o], inv = invL[o], g = gamL[o], bt = betL[o];
          #pragma unroll
          for (int r = 0; r < 8; ++r) {
            int lr2 = rt * 16 + r + 8 * hf;
            int nl2 = lr2 / K_;
            float h = (acc[ct][r] - mean) * inv * g + bt;
            h = (h >= 0.f) ? h : SLOPE_ * h;
            unsigned int u = __float_as_uint(h);
            u = (u & 0x80000000u) ? ~u : (u | 0x80000000u);  // monotone encode
            atomicMax(&mx[nl2 * O_ + o], u);
          }
        }
      }
    }
    __syncthreads();
  }

  if (STATS) {
    #pragma unroll
    for (int ct = 0; ct < 4; ++ct) {
      partS[wave][hf][ct * 16 + laneM] = s1a[ct];
      partQ[wave][hf][ct * 16 + laneM] = s2a[ct];
    }
    __syncthreads();
    if (tid < O_) {  // fixed-order block partial (deterministic)
      float s = partS[0][0][tid] + partS[0][1][tid] + partS[1][0][tid] + partS[1][1][tid];
      float q = partQ[0][0][tid] + partQ[0][1][tid] + partQ[1][0][tid] + partQ[1][1][tid];
      bpart[(size_t)blockIdx.x * 128 + tid]      = s;
      bpart[(size_t)blockIdx.x * 128 + 64 + tid] = q;
    }
  } else {
    for (int i = tid; i < NODES * O_; i += 64) {
      int nl = i >> 6, o = i & 63;
      unsigned int u = mx[i];
      u = (u & 0x80000000u) ? (u & 0x7fffffffu) : ~u;        // decode
      out[(size_t)b * O_ * N_ + (size_t)o * N_ + (n0 + nl)] = __uint_as_float(u);
    }
  }
}

// ------------------------------------------------------------------
// Kernel 3: deterministic reduction of block partials -> mean / invstd
// ------------------------------------------------------------------
__global__ void __launch_bounds__(128)
k_finalize(const float* __restrict__ part, float* __restrict__ stats) {
  __shared__ float red[128];
  int t = threadIdx.x;  // 0..127: [0,64) sum, [64,128) sumsq
  float s = 0.f;
  for (int blk = 0; blk < NBLK; ++blk) s += part[(size_t)blk * 128 + t];
  red[t] = s;
  __syncthreads();
  if (t < O_) {
    const float cnt = (float)((size_t)B_ * N_ * K_);
    float mean = red[t] / cnt;
    float var  = red[64 + t] / cnt - mean * mean;
    stats[t]      = mean;
    stats[64 + t] = rsqrtf(fmaxf(var, 0.f) + EPS_);
  }
}

// ------------------------------------------------------------------
extern "C" void kernel_launch(void* const* d_in, const int* in_sizes, int n_in,
                              void* d_out, int out_size, void* d_ws, size_t ws_size,
                              hipStream_t stream) {
  const float* x     = (const float*)d_in[0];
  const float* W     = (const float*)d_in[1];
  const float* gamma = (const float*)d_in[2];
  const float* beta  = (const float*)d_in[3];
  float* out = (float*)d_out;

  char* ws = (char*)d_ws;
  int*   idxb  = (int*)(ws + WS_IDX_OFF);
  float* xt    = (float*)(ws + WS_XT_OFF);
  float* sq    = (float*)(ws + WS_SQ_OFF);
  float* stats = (float*)(ws + WS_STAT_OFF);
  float* bpart = (float*)(ws + WS_PART_OFF);

  k_transpose<<<(B_ * N_ + 255) / 256, 256, 0, stream>>>(x, xt, sq);

  dim3 g1(N_ / ROWS, B_);
  k_knn<<<g1, 256, 0, stream>>>(xt, sq, idxb);

  k_edge<true><<<NBLK, 64, 0, stream>>>(xt, idxb, W, gamma, beta, bpart, stats, out);
  k_finalize<<<1, 128, 0, stream>>>(bpart, stats);
  k_edge<false><<<NBLK, 64, 0, stream>>>(xt, idxb, W, gamma, beta, bpart, stats, out);
}